// GINE_2757369004238
// MI455X (gfx1250) — compile-verified
//
#include <hip/hip_runtime.h>

typedef float v2f __attribute__((ext_vector_type(2)));
typedef float v8f __attribute__((ext_vector_type(8)));

#define BN_EPS 1e-5f

// ---------------------------------------------------------------------------
// Edge stage: one thread per edge.
//   e    = edge_attr[e] @ ew + eb            (8 x DIN, weights in LDS)
//   msg  = relu(x[src] + e)
//   agg[dst] += msg                          (hardware f32 atomics -> L2)
// ---------------------------------------------------------------------------
template<int DIN>
__global__ __launch_bounds__(256)
void gine_edge_kernel(const float* __restrict__ x,     // [N, DIN]
                      const int*   __restrict__ ei,    // [2, E]
                      const float* __restrict__ ea,    // [E, 8]
                      const float* __restrict__ ew,    // [8, DIN]
                      const float* __restrict__ ebias, // [DIN]
                      float*       __restrict__ agg,   // [N, DIN]
                      int E)
{
    __shared__ float s_w[9 * DIN];                     // ew (8*DIN) + eb (DIN)
    for (int i = threadIdx.x; i < 9 * DIN; i += 256)
        s_w[i] = (i < 8 * DIN) ? ew[i] : ebias[i - 8 * DIN];
    __syncthreads();

    int e = blockIdx.x * 256 + threadIdx.x;
    if (e >= E) return;

    const int src = ei[e];
    const int dst = ei[E + e];

    const float4 a0 = *(const float4*)(ea + (size_t)e * 8);
    const float4 a1 = *(const float4*)(ea + (size_t)e * 8 + 4);

    const float* __restrict__ xrow = x   + (size_t)src * DIN;
    float*       __restrict__ arow = agg + (size_t)dst * DIN;

    #pragma unroll
    for (int j4 = 0; j4 < DIN / 4; ++j4) {
        float4 xv = *(const float4*)(xrow + j4 * 4);
        #pragma unroll
        for (int u = 0; u < 4; ++u) {
            const int j = j4 * 4 + u;
            float acc = s_w[8 * DIN + j];              // eb[j]
            acc = fmaf(a0.x, s_w[0 * DIN + j], acc);
            acc = fmaf(a0.y, s_w[1 * DIN + j], acc);
            acc = fmaf(a0.z, s_w[2 * DIN + j], acc);
            acc = fmaf(a0.w, s_w[3 * DIN + j], acc);
            acc = fmaf(a1.x, s_w[4 * DIN + j], acc);
            acc = fmaf(a1.y, s_w[5 * DIN + j], acc);
            acc = fmaf(a1.z, s_w[6 * DIN + j], acc);
            acc = fmaf(a1.w, s_w[7 * DIN + j], acc);
            const float xj  = ((const float*)&xv)[u];
            const float msg = fmaxf(xj + acc, 0.0f);
            unsafeAtomicAdd(arow + j, msg);            // global_atomic_add_f32
        }
    }
}

// ---------------------------------------------------------------------------
// Node stage: one wave32 per 16-node tile, V_WMMA_F32_16X16X4_F32.
//   z = x + agg;  h = relu(BN(z @ wa + ba));  out = relu(h @ wb + bb)
// BN folded into GEMM1's B columns (scale) and post-accumulate bias.
// A(16x4 f32) per lane: M = lane&15, K = 4*kk + 2*(lane>=16) + r.
// C/D per lane: M = r + 8*(lane>=16), N = lane&15.
// D->A re-swizzle between the two GEMMs goes through a 16x17 LDS tile.
// ---------------------------------------------------------------------------
template<int DIN>
__global__ __launch_bounds__(256)
void gine_node_kernel(const float* __restrict__ x,    // [N, DIN]
                      const float* __restrict__ agg,  // [N, DIN]
                      const float* __restrict__ wa,   // [DIN, 16]
                      const float* __restrict__ ba,   // [16]
                      const float* __restrict__ g,
                      const float* __restrict__ be,
                      const float* __restrict__ bm,
                      const float* __restrict__ bv,
                      const float* __restrict__ wb,   // [16, 16]
                      const float* __restrict__ bb,   // [16]
                      float*       __restrict__ hout, // [N, 16]
                      int ntiles)
{
    __shared__ float s_h[8][16 * 17];                  // one 16x16 tile per wave
    const int wave = threadIdx.x >> 5;
    const int lane = threadIdx.x & 31;
    const int tile = blockIdx.x * 8 + wave;
    if (tile >= ntiles) return;                        // wave-uniform: EXEC stays ~0

    const int half = lane >> 4;                        // 0 | 1
    const int row  = lane & 15;                        // M for A, N for B/C/D
    const int m0   = tile * 16;
    const int mrow = m0 + row;

    // folded BatchNorm (eval)
    const float scale_n = g[row] * rsqrtf(bv[row] + BN_EPS);
    const float bias1_n = (ba[row] - bm[row]) * scale_n + be[row];

    const float* __restrict__ xrow = x   + (size_t)mrow * DIN;
    const float* __restrict__ arow = agg + (size_t)mrow * DIN;

    // ---- GEMM1: [16,DIN] @ [DIN,16], K-steps of 4 ----
    v8f c = {0.f, 0.f, 0.f, 0.f, 0.f, 0.f, 0.f, 0.f};
    #pragma unroll
    for (int kk = 0; kk < DIN / 4; ++kk) {
        const int kb = 4 * kk + 2 * half;
        v2f A = { xrow[kb] + arow[kb], xrow[kb + 1] + arow[kb + 1] };
        v2f B = { wa[kb * 16 + row] * scale_n, wa[(kb + 1) * 16 + row] * scale_n };
        c = __builtin_amdgcn_wmma_f32_16x16x4_f32(false, A, false, B,
                                                  (short)0, c, false, false);
    }

    // BN bias + ReLU, stage D tile into LDS in [m][k] order
    float* __restrict__ tl = &s_h[wave][0];
    #pragma unroll
    for (int r = 0; r < 8; ++r) {
        const float h = fmaxf(c[r] + bias1_n, 0.0f);
        tl[(r + 8 * half) * 17 + row] = h;
    }
    asm volatile("s_wait_dscnt 0" ::: "memory");       // within-wave LDS RAW

    // ---- GEMM2: [16,16] @ [16,16] ----
    v8f c2;
    const float b2 = bb[row];
    #pragma unroll
    for (int r = 0; r < 8; ++r) c2[r] = b2;
    #pragma unroll
    for (int kk = 0; kk < 4; ++kk) {
        const int kb = 4 * kk + 2 * half;
        v2f A = { tl[row * 17 + kb], tl[row * 17 + kb + 1] };
        v2f B = { wb[kb * 16 + row], wb[(kb + 1) * 16 + row] };
        c2 = __builtin_amdgcn_wmma_f32_16x16x4_f32(false, A, false, B,
                                                   (short)0, c2, false, false);
    }

    #pragma unroll
    for (int r = 0; r < 8; ++r) {
        const int mm = m0 + r + 8 * half;
        hout[(size_t)mm * 16 + row] = fmaxf(c2[r], 0.0f);
    }
}

// ---------------------------------------------------------------------------
// Final MLP: concat(h1,h2,h3)=[N,48] @ lw1[48,64] -> relu -> @ lw2[64,4].
// GEMM1 via WMMA (4 column tiles x 12 K-steps); 64->4 projection via VALU.
// ---------------------------------------------------------------------------
__global__ __launch_bounds__(256)
void gine_final_kernel(const float* __restrict__ h1,
                       const float* __restrict__ h2,
                       const float* __restrict__ h3,
                       const float* __restrict__ lw1,  // [48, 64]
                       const float* __restrict__ lb1,  // [64]
                       const float* __restrict__ lw2,  // [64, 4]
                       const float* __restrict__ lb2,  // [4]
                       float*       __restrict__ out,  // [N, 4]
                       int ntiles)
{
    __shared__ float s_hl[8][16 * 65];                 // 16x64 tile per wave, pad
    __shared__ float s_w2[64 * 4];
    __shared__ float s_b2[4];
    for (int i = threadIdx.x; i < 256; i += 256) s_w2[i] = lw2[i];
    if (threadIdx.x < 4) s_b2[threadIdx.x] = lb2[threadIdx.x];
    __syncthreads();

    const int wave = threadIdx.x >> 5;
    const int lane = threadIdx.x & 31;
    const int tile = blockIdx.x * 8 + wave;
    if (tile >= ntiles) return;

    const int half = lane >> 4;
    const int row  = lane & 15;
    const int m0   = tile * 16;
    const int mrow = m0 + row;

    const float* __restrict__ r1 = h1 + (size_t)mrow * 16;
    const float* __restrict__ r2 = h2 + (size_t)mrow * 16;
    const float* __restrict__ r3 = h3 + (size_t)mrow * 16;
    float* __restrict__ tl = &s_hl[wave][0];

    #pragma unroll
    for (int t = 0; t < 4; ++t) {                      // 4 column tiles of 16
        const int nc = t * 16 + row;
        v8f c = {0.f, 0.f, 0.f, 0.f, 0.f, 0.f, 0.f, 0.f};
        #pragma unroll
        for (int kk = 0; kk < 12; ++kk) {
            const int kb = 4 * kk + 2 * half;          // segment is wave-uniform
            float a0, a1;
            if (kb < 16)      { a0 = r1[kb];      a1 = r1[kb + 1];      }
            else if (kb < 32) { a0 = r2[kb - 16]; a1 = r2[kb - 15];     }
            else              { a0 = r3[kb - 32]; a1 = r3[kb - 31];     }
            v2f A = { a0, a1 };
            v2f B = { lw1[kb * 64 + nc], lw1[(kb + 1) * 64 + nc] };
            c = __builtin_amdgcn_wmma_f32_16x16x4_f32(false, A, false, B,
                                                      (short)0, c, false, false);
        }
        const float bias = lb1[nc];
        #pragma unroll
        for (int r = 0; r < 8; ++r)
            tl[(r + 8 * half) * 65 + nc] = fmaxf(c[r] + bias, 0.0f);
    }
    asm volatile("s_wait_dscnt 0" ::: "memory");

    // 64 -> 4 projection: lane handles (m=row, o = 2*half + {0,1})
    #pragma unroll
    for (int oo = 0; oo < 2; ++oo) {
        const int o = 2 * half + oo;
        float acc = s_b2[o];
        #pragma unroll
        for (int k = 0; k < 64; ++k)
            acc = fmaf(tl[row * 65 + k], s_w2[k * 4 + o], acc);
        out[(size_t)(m0 + row) * 4 + o] = acc;
    }
}

// ---------------------------------------------------------------------------
extern "C" void kernel_launch(void* const* d_in, const int* in_sizes, int n_in,
                              void* d_out, int out_size, void* d_ws, size_t ws_size,
                              hipStream_t stream)
{
    const float* x  = (const float*)d_in[0];
    const int*   ei = (const int*)  d_in[1];
    const float* ea = (const float*)d_in[2];
    // per-layer params: base = 3 + 10*(c-1): ew eb wa ba g be m v wb bb
    #define LP(c, k) ((const float*)d_in[3 + 10 * ((c) - 1) + (k)])
    const float* lw1 = (const float*)d_in[33];
    const float* lb1 = (const float*)d_in[34];
    const float* lw2 = (const float*)d_in[35];
    const float* lb2 = (const float*)d_in[36];

    const int N = in_sizes[0] / 32;
    const int E = in_sizes[1] / 2;
    const int ntiles = N / 16;                         // N = 100000 -> 6250 tiles

    float* agg = (float*)d_ws;                         // N*32 floats (max din)
    float* h1  = agg + (size_t)N * 32;                 // N*16 each
    float* h2  = h1 + (size_t)N * 16;
    float* h3  = h2 + (size_t)N * 16;

    const dim3 blk(256);
    const dim3 egrid((E + 255) / 256);
    const dim3 ngrid((ntiles + 7) / 8);

    // ---- layer 1 (din = 32) ----
    hipMemsetAsync(agg, 0, (size_t)N * 32 * sizeof(float), stream);
    gine_edge_kernel<32><<<egrid, blk, 0, stream>>>(x, ei, ea, LP(1,0), LP(1,1), agg, E);
    gine_node_kernel<32><<<ngrid, blk, 0, stream>>>(x, agg, LP(1,2), LP(1,3), LP(1,4),
                                                    LP(1,5), LP(1,6), LP(1,7),
                                                    LP(1,8), LP(1,9), h1, ntiles);
    // ---- layer 2 (din = 16) ----
    hipMemsetAsync(agg, 0, (size_t)N * 16 * sizeof(float), stream);
    gine_edge_kernel<16><<<egrid, blk, 0, stream>>>(h1, ei, ea, LP(2,0), LP(2,1), agg, E);
    gine_node_kernel<16><<<ngrid, blk, 0, stream>>>(h1, agg, LP(2,2), LP(2,3), LP(2,4),
                                                    LP(2,5), LP(2,6), LP(2,7),
                                                    LP(2,8), LP(2,9), h2, ntiles);
    // ---- layer 3 (din = 16) ----
    hipMemsetAsync(agg, 0, (size_t)N * 16 * sizeof(float), stream);
    gine_edge_kernel<16><<<egrid, blk, 0, stream>>>(h2, ei, ea, LP(3,0), LP(3,1), agg, E);
    gine_node_kernel<16><<<ngrid, blk, 0, stream>>>(h2, agg, LP(3,2), LP(3,3), LP(3,4),
                                                    LP(3,5), LP(3,6), LP(3,7),
                                                    LP(3,8), LP(3,9), h3, ntiles);
    // ---- final MLP ----
    gine_final_kernel<<<ngrid, blk, 0, stream>>>(h1, h2, h3, lw1, lb1, lw2, lb2,
                                                 (float*)d_out, ntiles);
    #undef LP
}